// GCNEncoder_69166153334987
// MI455X (gfx1250) — compile-verified
//
#include <hip/hip_runtime.h>

typedef __attribute__((ext_vector_type(2))) float v2f;
typedef __attribute__((ext_vector_type(8))) float v8f;

// ---------------------------------------------------------------------------
// Utility kernels
// ---------------------------------------------------------------------------
__global__ __launch_bounds__(256) void zero_f32(float* __restrict__ p, long long n) {
  long long i = (long long)blockIdx.x * blockDim.x + threadIdx.x;
  if (i < n) p[i] = 0.0f;
}

__global__ __launch_bounds__(256) void degree_kernel(const int* __restrict__ dst,
                                                     float* __restrict__ deg, int nE) {
  int e = blockIdx.x * blockDim.x + threadIdx.x;
  if (e < nE) atomicAdd(&deg[dst[e]], 1.0f);
}

__global__ __launch_bounds__(256) void invdeg_kernel(const float* __restrict__ deg,
                                                     float* __restrict__ invd, int n) {
  int i = blockIdx.x * blockDim.x + threadIdx.x;
  if (i < n) invd[i] = 1.0f / fmaxf(deg[i], 1.0f);
}

// One thread per (edge, channel). ch is a power of two -> shift/mask addressing.
__global__ __launch_bounds__(256) void scatter_add_kernel(
    const float* __restrict__ feat, const int* __restrict__ src,
    const int* __restrict__ dst, float* __restrict__ agg,
    long long total, int lc, int chmask) {
  long long tid = (long long)blockIdx.x * blockDim.x + threadIdx.x;
  if (tid >= total) return;
  int e = (int)(tid >> lc);
  int c = (int)tid & chmask;
  int s = src[e];
  int d = dst[e];
  long long ch = (long long)(chmask + 1);
  atomicAdd(&agg[(long long)d * ch + c], feat[(long long)s * ch + c]);
}

// ---------------------------------------------------------------------------
// Fused SAGE GEMM:  out = act( (agg * invd) @ Wl^T + Xs @ Wr^T + bias )
//
// One wave computes a 16(M) x 32(N) output tile with V_WMMA_F32_16X16X4_F32:
// two independent accumulator chains (c0, c1) share every A-fragment, halving
// the streaming X/agg load traffic and letting the scheduler overlap one
// chain's wmma issue with the other chain's fragment loads.
//
// Fragment layouts (f32 16x16x4, wave32):
//  A (16x4): lane L -> M = L&15; float2 at col k + ((L>>4)<<1)
//  B (4x16): lane L -> N = L&15 of W[DOUT][DIN]; same cols
//  C (16x16, v8f): vgpr i, lanes 0-15 -> (M=i, N=lane); 16-31 -> (M=i+8)
// Bias is folded into the accumulator init (N depends only on lane).
// ---------------------------------------------------------------------------
template <bool DUAL, bool RELU>
__global__ __launch_bounds__(256) void sage_gemm_kernel(
    const float* __restrict__ Xs, int ldxs,
    const float* __restrict__ Xa, int ldxa,
    const float* __restrict__ invd,
    const float* __restrict__ Wl,    // [DOUT, kdim] (agg path)
    const float* __restrict__ Wr,    // [DOUT, kdim] (self path)
    const float* __restrict__ bias,  // [DOUT]
    float* __restrict__ out, int ldo,
    int npairs, int total_tiles, int kdim) {
  int wave = blockIdx.x * (blockDim.x >> 5) + (threadIdx.x >> 5);
  if (wave >= total_tiles) return;  // whole-wave exit: EXEC stays all-ones
  int np = wave % npairs;
  int mt = wave / npairs;

  int lane  = threadIdx.x & 31;
  int lrow  = lane & 15;             // M for A-frags, N for B-frags
  int khalf = (lane >> 4) << 1;      // 0 for lanes 0-15, 2 for lanes 16-31
  int m0 = mt << 4;
  int n0 = np << 5;                  // 32-wide N tile

  const float* xs_row  = Xs + (long long)(m0 + lrow) * ldxs + khalf;
  const float* wr0_row = Wr + (long long)(n0 + lrow) * kdim + khalf;
  const float* wr1_row = Wr + (long long)(n0 + 16 + lrow) * kdim + khalf;
  const float* xa_row  = nullptr;
  const float* wl0_row = nullptr;
  const float* wl1_row = nullptr;
  float s = 0.0f;
  if (DUAL) {
    xa_row  = Xa + (long long)(m0 + lrow) * ldxa + khalf;
    wl0_row = Wl + (long long)(n0 + lrow) * kdim + khalf;
    wl1_row = Wl + (long long)(n0 + 16 + lrow) * kdim + khalf;
    s = invd[m0 + lrow];
  }

  // Bias folded into accumulator init: every C element in this lane has
  // N = n0 + lrow (chain 0) or N = n0 + 16 + lrow (chain 1).
  float b0 = bias[n0 + lrow];
  float b1 = bias[n0 + 16 + lrow];
  v8f c0, c1;
#pragma unroll
  for (int i = 0; i < 8; ++i) { c0[i] = b0; c1[i] = b1; }

  for (int k = 0; k < kdim; k += 4) {
    v2f as  = *reinterpret_cast<const v2f*>(xs_row + k);
    v2f br0 = *reinterpret_cast<const v2f*>(wr0_row + k);
    v2f br1 = *reinterpret_cast<const v2f*>(wr1_row + k);
    c0 = __builtin_amdgcn_wmma_f32_16x16x4_f32(false, as, false, br0,
                                               (short)0, c0, false, false);
    c1 = __builtin_amdgcn_wmma_f32_16x16x4_f32(false, as, false, br1,
                                               (short)0, c1, false, false);
    if (DUAL) {
      v2f aa = *reinterpret_cast<const v2f*>(xa_row + k);
      aa.x *= s;
      aa.y *= s;
      v2f bl0 = *reinterpret_cast<const v2f*>(wl0_row + k);
      v2f bl1 = *reinterpret_cast<const v2f*>(wl1_row + k);
      c0 = __builtin_amdgcn_wmma_f32_16x16x4_f32(false, aa, false, bl0,
                                                 (short)0, c0, false, false);
      c1 = __builtin_amdgcn_wmma_f32_16x16x4_f32(false, aa, false, bl1,
                                                 (short)0, c1, false, false);
    }
  }

  float* orow = out + (long long)(m0 + ((lane >> 4) << 3)) * ldo + n0 + lrow;
#pragma unroll
  for (int i = 0; i < 8; ++i) {
    float v0 = c0[i];
    float v1 = c1[i];
    if (RELU) {
      v0 = fmaxf(v0, 0.0f);
      v1 = fmaxf(v1, 0.0f);
    }
    orow[(long long)i * ldo] = v0;
    orow[(long long)i * ldo + 16] = v1;
  }
}

// ---------------------------------------------------------------------------
// Host-side orchestration
// ---------------------------------------------------------------------------
static inline unsigned cdiv_ll(long long a, long long b) {
  return (unsigned)((a + b - 1) / b);
}
static inline int ilog2i(int v) { return 31 - __builtin_clz((unsigned)v); }

extern "C" void kernel_launch(void* const* d_in, const int* in_sizes, int n_in,
                              void* d_out, int out_size, void* d_ws, size_t ws_size,
                              hipStream_t stream) {
  const float* x    = (const float*)d_in[0];
  const int*   ei   = (const int*)d_in[1];
  const float* Wl1  = (const float*)d_in[2];
  const float* bl1  = (const float*)d_in[3];
  const float* Wr1  = (const float*)d_in[4];
  const float* Wl2  = (const float*)d_in[5];
  const float* bl2  = (const float*)d_in[6];
  const float* Wr2  = (const float*)d_in[7];
  const float* Wl4  = (const float*)d_in[8];
  const float* bl4  = (const float*)d_in[9];
  const float* Wr4  = (const float*)d_in[10];
  const float* Wlin = (const float*)d_in[11];
  const float* blin = (const float*)d_in[12];

  const int d1   = in_sizes[3];            // 256
  const int IN   = in_sizes[2] / d1;       // 128
  const int d2   = in_sizes[6];            // 128
  const int d4   = in_sizes[9];            // 64
  const int dlin = in_sizes[11] / d4;      // Linear input width (derived)
  const int N    = in_sizes[0] / IN;       // 50000
  const int E    = in_sizes[1] / 2;        // 800000
  const int* srcI = ei;
  const int* dstI = ei + E;

  // Workspace layout (floats)
  float* ws   = (float*)d_ws;
  float* deg  = ws;
  float* invd = deg + N;
  float* agg  = invd + N;                  // N x d1 (max width)
  float* h1   = agg + (long long)N * d1;   // N x d1
  float* h2   = h1 + (long long)N * d1;    // N x d2
  float* h3   = h1;                        // reuse: h1 dead after conv2

  const int B = 256;                       // 8 waves per block
  const int mtiles = N / 16;               // 50000/16 = 3125, exact

  // Degree + inverse degree
  zero_f32<<<cdiv_ll(N, B), B, 0, stream>>>(deg, N);
  degree_kernel<<<cdiv_ll(E, B), B, 0, stream>>>(dstI, deg, E);
  invdeg_kernel<<<cdiv_ll(N, B), B, 0, stream>>>(deg, invd, N);

  // ---- conv1: agg(x) -> h1 = relu(agg@Wl1^T + bl1 + x@Wr1^T) --------------
  {
    long long aggN = (long long)N * IN;
    zero_f32<<<cdiv_ll(aggN, B), B, 0, stream>>>(agg, aggN);
    long long tot = (long long)E * IN;
    scatter_add_kernel<<<cdiv_ll(tot, B), B, 0, stream>>>(
        x, srcI, dstI, agg, tot, ilog2i(IN), IN - 1);
    int npairs = d1 / 32, total = mtiles * npairs;
    sage_gemm_kernel<true, true><<<cdiv_ll(total, 8), B, 0, stream>>>(
        x, IN, agg, IN, invd, Wl1, Wr1, bl1, h1, d1, npairs, total, IN);
  }

  // ---- conv2: agg(h1) -> h2 = relu(agg@Wl2^T + bl2 + h1@Wr2^T) ------------
  {
    long long aggN = (long long)N * d1;
    zero_f32<<<cdiv_ll(aggN, B), B, 0, stream>>>(agg, aggN);
    long long tot = (long long)E * d1;
    scatter_add_kernel<<<cdiv_ll(tot, B), B, 0, stream>>>(
        h1, srcI, dstI, agg, tot, ilog2i(d1), d1 - 1);
    int npairs = d2 / 32, total = mtiles * npairs;
    sage_gemm_kernel<true, true><<<cdiv_ll(total, 8), B, 0, stream>>>(
        h1, d1, agg, d1, invd, Wl2, Wr2, bl2, h2, d2, npairs, total, d1);
  }

  // ---- linear: h3 = h2 @ Wlin^T + blin (dense, no agg, no relu) -----------
  {
    int npairs = d4 / 32, total = mtiles * npairs;
    sage_gemm_kernel<false, false><<<cdiv_ll(total, 8), B, 0, stream>>>(
        h2, d2, nullptr, 0, nullptr, nullptr, Wlin, blin, h3, d4,
        npairs, total, dlin);
  }

  // ---- conv4: agg(h3) -> out = agg@Wl4^T + bl4 + h3@Wr4^T -----------------
  {
    long long aggN = (long long)N * d4;
    zero_f32<<<cdiv_ll(aggN, B), B, 0, stream>>>(agg, aggN);
    long long tot = (long long)E * d4;
    scatter_add_kernel<<<cdiv_ll(tot, B), B, 0, stream>>>(
        h3, srcI, dstI, agg, tot, ilog2i(d4), d4 - 1);
    int npairs = d4 / 32, total = mtiles * npairs;
    sage_gemm_kernel<true, false><<<cdiv_ll(total, 8), B, 0, stream>>>(
        h3, d4, agg, d4, invd, Wl4, Wr4, bl4, (float*)d_out, d4,
        npairs, total, d4);
  }
}